// SubgraphEmbeddingRegressorModel_87737591923132
// MI455X (gfx1250) — compile-verified
//
#include <hip/hip_runtime.h>
#include <hip/hip_bf16.h>
#include <stdint.h>

typedef __attribute__((ext_vector_type(2))) float v2f;
typedef __attribute__((ext_vector_type(8))) float v8f;

#define NN 500000   // nodes
#define NE 2000000  // edges
#define NG 20000    // graphs
#define NP 50000    // drug pairs

// ---------------------------------------------------------------- utilities
__global__ void k_fill(float* __restrict__ p, float v, int n) {
  int i = blockIdx.x * blockDim.x + threadIdx.x;
  if (i < n) p[i] = v;
}

// deg[dst] += 1 per edge (deg pre-filled with 1.0 for the self loop)
__global__ void k_degree(const int* __restrict__ dst, float* __restrict__ deg, int E) {
  int e = blockIdx.x * blockDim.x + threadIdx.x;
  if (e < E) unsafeAtomicAdd(&deg[dst[e]], 1.0f);
}

// ---------------------------------------------------------------- WMMA GEMM
// Y[M,N] = X[M,K] @ W[K,N]  (+bias, +relu optional), fp32 via V_WMMA_F32_16X16X4_F32.
// One wave per 16x16 output tile; M and N must be multiples of 16.
template <int K, int N>
__global__ void k_gemm_wmma(const float* __restrict__ X, const float* __restrict__ W,
                            const float* __restrict__ bias, float* __restrict__ Y,
                            int M, int fuse_relu) {
  const int lane = threadIdx.x & 31;
  const int tilesN = N / 16;
  const int tile = blockIdx.x * (blockDim.x >> 5) + (threadIdx.x >> 5);
  const int totalTiles = (M / 16) * tilesN;
  if (tile >= totalTiles) return;  // wave-uniform: EXEC all-ones for WMMA

  const int tm = tile / tilesN;
  const int tn = tile % tilesN;
  const size_t row0 = (size_t)tm * 16;
  const int col0 = tn * 16;
  const int mn = lane & 15;        // A: row in tile; B: col in tile
  const int kb = (lane >> 4) * 2;  // K sub-offset per half-wave

  v8f acc = {};
#pragma unroll
  for (int k0 = 0; k0 < K; k0 += 4) {
    v2f a, b;
    const float* xr = X + (row0 + mn) * (size_t)K + (size_t)(k0 + kb);
    a.x = xr[0];
    a.y = xr[1];
    const float* wr = W + (size_t)(k0 + kb) * N + (size_t)(col0 + mn);
    b.x = wr[0];
    b.y = wr[N];
    acc = __builtin_amdgcn_wmma_f32_16x16x4_f32(false, a, false, b, (short)0, acc,
                                                false, false);
  }

  const int rbase = (lane >> 4) * 8;  // C/D layout: lanes 16-31 hold rows +8
#pragma unroll
  for (int r = 0; r < 8; ++r) {
    size_t row = row0 + (size_t)(rbase + r);
    float v = acc[r];
    if (bias) v += bias[col0 + mn];
    if (fuse_relu) v = fmaxf(v, 0.0f);
    Y[row * (size_t)N + (size_t)(col0 + mn)] = v;
  }
}

// ---------------------------------------------------------------- GCN pieces
// agg[dst,c] += h[src,c] * rsqrt(deg[src]) * rsqrt(deg[dst])
__global__ void k_scatter(const float* __restrict__ H, const float* __restrict__ deg,
                          const int* __restrict__ src, const int* __restrict__ dst,
                          float* __restrict__ agg, int E) {
  long long idx = (long long)blockIdx.x * blockDim.x + threadIdx.x;
  int e = (int)(idx >> 6);
  int c = (int)(idx & 63);
  if (e >= E) return;
  int s = src[e];
  int d = dst[e];
  float nrm = rsqrtf(deg[s]) * rsqrtf(deg[d]);
  unsafeAtomicAdd(&agg[(size_t)d * 64 + c], H[(size_t)s * 64 + c] * nrm);
}

// out = relu(agg + h/deg + b)
__global__ void k_post(const float* __restrict__ agg, const float* __restrict__ H,
                       const float* __restrict__ deg, const float* __restrict__ b,
                       float* __restrict__ out, long long n64) {
  long long idx = (long long)blockIdx.x * blockDim.x + threadIdx.x;
  if (idx >= n64) return;
  int node = (int)(idx >> 6);
  int c = (int)(idx & 63);
  float v = agg[idx] + H[idx] * (1.0f / deg[node]) + b[c];
  out[idx] = fmaxf(v, 0.0f);
}

// sums[batch[n],c] += h[n,c];  counts[batch[n]] += 1
__global__ void k_pool(const float* __restrict__ H, const int* __restrict__ batch,
                       float* __restrict__ sums, float* __restrict__ counts,
                       long long n64) {
  long long idx = (long long)blockIdx.x * blockDim.x + threadIdx.x;
  if (idx >= n64) return;
  int node = (int)(idx >> 6);
  int c = (int)(idx & 63);
  int g = batch[node];
  unsafeAtomicAdd(&sums[(size_t)g * 64 + c], H[idx]);
  if (c == 0) unsafeAtomicAdd(&counts[g], 1.0f);
}

__global__ void k_graph_mean(const float* __restrict__ sums, const float* __restrict__ counts,
                             float* __restrict__ ge, int g64) {
  int idx = blockIdx.x * blockDim.x + threadIdx.x;
  if (idx >= g64) return;
  int g = idx >> 6;
  ge[idx] = sums[idx] / fmaxf(counts[g], 1.0f);
}

// feat[p, 0:64] = ge[pairs[0,p]];  feat[p, 64:128] = ge[pairs[1,p]]
__global__ void k_pairfeat(const float* __restrict__ ge, const int* __restrict__ pairs,
                           float* __restrict__ feat, int P) {
  long long idx = (long long)blockIdx.x * blockDim.x + threadIdx.x;
  int p = (int)(idx >> 7);
  int c = (int)(idx & 127);
  if (p >= P) return;
  int g = (c < 64) ? pairs[p] : pairs[(size_t)P + p];
  feat[idx] = ge[(size_t)g * 64 + (c & 63)];
}

// out[p] = dot(hid[p,:256], Wr2) + br2   (one wave32 per pair)
__global__ void k_head(const float* __restrict__ hid, const float* __restrict__ w,
                       const float* __restrict__ b, float* __restrict__ out, int P) {
  int wave = (blockIdx.x * blockDim.x + threadIdx.x) >> 5;
  int lane = threadIdx.x & 31;
  if (wave >= P) return;
  const float* h = hid + (size_t)wave * 256;
  float s = 0.0f;
#pragma unroll
  for (int j = 0; j < 8; ++j) s += h[lane + 32 * j] * w[lane + 32 * j];
#pragma unroll
  for (int off = 16; off > 0; off >>= 1) s += __shfl_xor(s, off, 32);
  if (lane == 0) out[wave] = s + b[0];
}

// ---------------------------------------------------------------- launcher
extern "C" void kernel_launch(void* const* d_in, const int* in_sizes, int n_in,
                              void* d_out, int out_size, void* d_ws, size_t ws_size,
                              hipStream_t stream) {
  const float* x    = (const float*)d_in[0];
  const int*   edge = (const int*)d_in[1];   // [2, NE]: src = edge, dst = edge+NE
  const int*   batch = (const int*)d_in[2];
  const int*   pairs = (const int*)d_in[3];  // [2, NP]
  const float* W1  = (const float*)d_in[4];
  const float* b1  = (const float*)d_in[5];
  const float* W2  = (const float*)d_in[6];
  const float* b2  = (const float*)d_in[7];
  const float* Wr1 = (const float*)d_in[8];
  const float* br1 = (const float*)d_in[9];
  const float* Wr2 = (const float*)d_in[10];
  const float* br2 = (const float*)d_in[11];
  float* out = (float*)d_out;

  char* ws = (char*)d_ws;
  size_t off = 0;
  auto alloc = [&](size_t bytes) -> float* {
    float* p = (float*)(ws + off);
    off += (bytes + 255) & ~(size_t)255;
    return p;
  };
  float* deg    = alloc((size_t)NN * 4);
  float* bufH   = alloc((size_t)NN * 64 * 4);  // x@W (raw)
  float* bufAGG = alloc((size_t)NN * 64 * 4);  // scatter accumulator
  float* bufACT = alloc((size_t)NN * 64 * 4);  // post-activation
  float* sums   = alloc((size_t)NG * 64 * 4);
  float* counts = alloc((size_t)NG * 4);
  float* ge     = alloc((size_t)NG * 64 * 4);
  float* feat = bufH;    // reuse after layer-2 GEMM consumed
  float* hid  = bufAGG;  // reuse after layer-2 post consumed

  const int* src = edge;
  const int* dst = edge + NE;

  const long long n64 = (long long)NN * 64;
  const int blk = 256;
  const int gN64 = (int)((n64 + blk - 1) / blk);
  const int gE64 = (int)(((long long)NE * 64 + blk - 1) / blk);

  // degrees (with self loop)
  k_fill<<<(NN + blk - 1) / blk, blk, 0, stream>>>(deg, 1.0f, NN);
  k_degree<<<(NE + blk - 1) / blk, blk, 0, stream>>>(dst, deg, NE);

  // ----- GCN layer 1
  {
    int tiles = (NN / 16) * (64 / 16);
    k_gemm_wmma<64, 64><<<(tiles + 7) / 8, blk, 0, stream>>>(x, W1, nullptr, bufH, NN, 0);
  }
  k_fill<<<gN64, blk, 0, stream>>>(bufAGG, 0.0f, (int)n64);
  k_scatter<<<gE64, blk, 0, stream>>>(bufH, deg, src, dst, bufAGG, NE);
  k_post<<<gN64, blk, 0, stream>>>(bufAGG, bufH, deg, b1, bufACT, n64);

  // ----- GCN layer 2
  {
    int tiles = (NN / 16) * (64 / 16);
    k_gemm_wmma<64, 64><<<(tiles + 7) / 8, blk, 0, stream>>>(bufACT, W2, nullptr, bufH, NN, 0);
  }
  k_fill<<<gN64, blk, 0, stream>>>(bufAGG, 0.0f, (int)n64);
  k_scatter<<<gE64, blk, 0, stream>>>(bufH, deg, src, dst, bufAGG, NE);
  k_post<<<gN64, blk, 0, stream>>>(bufAGG, bufH, deg, b2, bufACT, n64);

  // ----- scatter-mean pooling
  k_fill<<<(NG * 64 + blk - 1) / blk, blk, 0, stream>>>(sums, 0.0f, NG * 64);
  k_fill<<<(NG + blk - 1) / blk, blk, 0, stream>>>(counts, 0.0f, NG);
  k_pool<<<gN64, blk, 0, stream>>>(bufACT, batch, sums, counts, n64);
  k_graph_mean<<<(NG * 64 + blk - 1) / blk, blk, 0, stream>>>(sums, counts, ge, NG * 64);

  // ----- pair MLP head
  k_pairfeat<<<(int)(((long long)NP * 128 + blk - 1) / blk), blk, 0, stream>>>(ge, pairs, feat, NP);
  {
    int tiles = (NP / 16) * (256 / 16);
    k_gemm_wmma<128, 256><<<(tiles + 7) / 8, blk, 0, stream>>>(feat, Wr1, br1, hid, NP, 1);
  }
  k_head<<<(NP * 32 + blk - 1) / blk, blk, 0, stream>>>(hid, Wr2, br2, out, NP);
}